// model_70317204570866
// MI455X (gfx1250) — compile-verified
//
#include <hip/hip_runtime.h>
#include <hip/hip_bf16.h>

#define EPSF       1e-8f
#define NUM_ITEM   100000
#define DIM        256
#define NUM_SETS   8192
#define SET_LEN    50
#define BATCH      131072

typedef float v2f __attribute__((ext_vector_type(2)));
typedef float v8f __attribute__((ext_vector_type(8)));

__device__ __forceinline__ float softplus1(float x) {
    // log(1 + exp(x)), numerically stable, BETA == 1
    if (x > 20.f)  return x;
    if (x < -20.f) return __expf(x);
    return log1pf(__expf(x));
}

__device__ __forceinline__ float wave_sum(float v) {
    #pragma unroll
    for (int off = 16; off; off >>= 1) v += __shfl_xor(v, off, 32);
    return v;
}
__device__ __forceinline__ float wave_max(float v) {
    #pragma unroll
    for (int off = 16; off; off >>= 1) v = fmaxf(v, __shfl_xor(v, off, 32));
    return v;
}

// ---------------------------------------------------------------------------
// Kernel 1: att = X @ A  (100000x256 GEMV, f32) via V_WMMA_F32_16X16X4_F32.
// One wave per 16-row tile. B holds the shared vector A replicated across the
// 16 N-columns, so D[m, n] == att[row_base + m] for every n.
// A-matrix 16x4 f32 layout: lanes 0-15 -> M=lane, VGPR0/1 = K=0/1;
//                           lanes 16-31 -> M=lane-16, VGPR0/1 = K=2/3.
// D 16x16 f32 layout: VGPR v: lanes 0-15 -> M=v, lanes 16-31 -> M=v+8.
// Memory-bound (102 MB read @ 23.3 TB/s), so the N-replication is free.
// ---------------------------------------------------------------------------
__global__ void attn_matvec_wmma(const float* __restrict__ X,
                                 const float* __restrict__ A,
                                 float* __restrict__ att, int nrows) {
    const int wave = (blockIdx.x * blockDim.x + threadIdx.x) >> 5;
    const int lane = threadIdx.x & 31;
    const int row_base = wave * 16;
    if (row_base >= nrows) return;            // wave-uniform: EXEC stays all-1s

    const int m  = lane & 15;
    const int k0 = (lane >> 4) << 1;          // 0 for lanes 0-15, 2 for 16-31
    const float* __restrict__ rowp = X + (size_t)(row_base + m) * DIM;

    v8f acc = {};
    #pragma unroll 4
    for (int kb = 0; kb < DIM; kb += 4) {
        v2f a, b;
        a.x = rowp[kb + k0];
        a.y = rowp[kb + k0 + 1];
        b.x = A[kb + k0];                     // shared vector, replicated over N
        b.y = A[kb + k0 + 1];
        acc = __builtin_amdgcn_wmma_f32_16x16x4_f32(
                  false, a, false, b, (short)0, acc, false, false);
    }
    // Column n=0 lives in lane 0 (M=0..7) and lane 16 (M=8..15).
    if (m == 0) {
        const int rb = row_base + ((lane >> 4) << 3);
        #pragma unroll
        for (int v = 0; v < 8; ++v) att[rb + v] = acc[v];
    }
}

// ---------------------------------------------------------------------------
// Kernel 2: per-set masked softmax attention + weighted embeddings for both
// tables, plus the per-set radius box-volume (hoisted out of the batch loop:
// bv[s] = sum_d log(softplus(emb_r[s,d]) + eps)).
// One 256-thread block per set; wave0/wave1 do the two softmaxes, then each
// thread owns one of the 256 dims for the gather-weighted sum (coalesced).
// ---------------------------------------------------------------------------
__global__ void set_embed_kernel(const int* __restrict__ S,
                                 const int* __restrict__ Mm,
                                 const float* __restrict__ Xc,
                                 const float* __restrict__ Xr,
                                 const float* __restrict__ att_c,
                                 const float* __restrict__ att_r,
                                 float* __restrict__ emb_c,
                                 float* __restrict__ emb_r,
                                 float* __restrict__ bv) {
    __shared__ float wc[SET_LEN];
    __shared__ float wr[SET_LEN];
    __shared__ int   idx[SET_LEN];
    __shared__ float s_scale;
    __shared__ float red[8];

    const int s    = blockIdx.x;
    const int tid  = threadIdx.x;
    const int w    = tid >> 5;
    const int lane = tid & 31;

    if (w < 2) {
        const float* att = (w == 0) ? att_c : att_r;
        const int l0 = lane, l1 = lane + 32;
        const int  id0 = S[s * SET_LEN + l0];
        const int  mk0 = Mm[s * SET_LEN + l0];
        int id1 = 0, mk1 = 0;
        if (l1 < SET_LEN) { id1 = S[s * SET_LEN + l1]; mk1 = Mm[s * SET_LEN + l1]; }
        const float lg0 = mk0 ? att[id0] : -1e30f;
        const float lg1 = (l1 < SET_LEN && mk1) ? att[id1] : -1e30f;
        if (w == 0) { idx[l0] = id0; if (l1 < SET_LEN) idx[l1] = id1; }

        const float mx = wave_max(fmaxf(lg0, lg1));
        const float e0 = mk0 ? __expf(lg0 - mx) : 0.f;
        const float e1 = (l1 < SET_LEN && mk1) ? __expf(lg1 - mx) : 0.f;
        float sm = wave_sum(e0 + e1);
        sm = fmaxf(sm, 1e-30f);

        float* wdst = (w == 0) ? wc : wr;
        wdst[l0] = e0 / sm;
        if (l1 < SET_LEN) wdst[l1] = e1 / sm;

        if (w == 0) {
            const float cnt = wave_sum((float)(mk0 + ((l1 < SET_LEN) ? mk1 : 0)));
            if (lane == 0) s_scale = powf(cnt, 1.0f / (float)DIM);
        }
    }
    __syncthreads();

    const int d = tid;                        // one dim per thread (coalesced)
    float accc = 0.f, accr = 0.f;
    #pragma unroll 5
    for (int l = 0; l < SET_LEN; ++l) {
        const size_t base = (size_t)idx[l] * DIM + d;
        accc = fmaf(wc[l], Xc[base], accc);
        accr = fmaf(wr[l], Xr[base], accr);
    }
    accc *= s_scale;
    accr *= s_scale;
    emb_c[(size_t)s * DIM + d] = accc;
    emb_r[(size_t)s * DIM + d] = accr;

    // per-set radius box volume
    float t = wave_sum(__logf(softplus1(accr) + EPSF));
    if (lane == 0) red[w] = t;
    __syncthreads();
    if (tid == 0) {
        float acc = 0.f;
        #pragma unroll
        for (int x = 0; x < 8; ++x) acc += red[x];
        bv[s] = acc;
    }
}

// ---------------------------------------------------------------------------
// Kernel 3: 131072 pairwise box losses. One wave per pair, 8 dims/lane,
// 128B-coalesced passes over the L2-resident 8MB embedding tables; shfl
// reduction; block-aggregated float atomics into the 4 scalar losses.
// ---------------------------------------------------------------------------
__global__ void batch_loss_kernel(const int* __restrict__ inst,
                                  const float* __restrict__ sims,
                                  const float* __restrict__ emb_c,
                                  const float* __restrict__ emb_r,
                                  const float* __restrict__ bv,
                                  float* __restrict__ out) {
    __shared__ float part[4];
    const int tid  = threadIdx.x;
    const int w    = tid >> 5;
    const int lane = tid & 31;
    if (tid < 4) part[tid] = 0.f;
    __syncthreads();

    const int b = blockIdx.x * 8 + w;
    const int i = inst[2 * b + 0];
    const int j = inst[2 * b + 1];
    const float* __restrict__ ci = emb_c + (size_t)i * DIM;
    const float* __restrict__ cj = emb_c + (size_t)j * DIM;
    const float* __restrict__ ri = emb_r + (size_t)i * DIM;
    const float* __restrict__ rj = emb_r + (size_t)j * DIM;

    float si = 0.f, su = 0.f;
    #pragma unroll
    for (int q = 0; q < DIM / 32; ++q) {
        const int d = lane + q * 32;
        const float a  = ci[d];
        const float c2 = cj[d];
        const float Mi = a  + ri[d];
        const float Mj = c2 + rj[d];
        const float xin = fminf(Mi, Mj) - fmaxf(a, c2);
        const float xun = fmaxf(Mi, Mj) - fminf(a, c2);
        si += __logf(softplus1(xin) + EPSF);
        su += __logf(softplus1(xun) + EPSF);
    }
    si = wave_sum(si);
    su = wave_sum(su);

    if (lane == 0) {
        const float bi = bv[i], bj = bv[j];
        const float c_ovl = __expf(si - fmaxf(bi, bj));
        const float c_jac = __expf(si - su);
        const float c_cos = __expf(si - 0.5f * (bi + bj));
        const float c_dic = 2.f * __expf(si) / (__expf(bi) + __expf(bj) + EPSF);
        const float4 sv = *(const float4*)(sims + 4 * (size_t)b);
        atomicAdd(&part[0], (c_ovl - sv.x) * (c_ovl - sv.x));
        atomicAdd(&part[1], (c_jac - sv.y) * (c_jac - sv.y));
        atomicAdd(&part[2], (c_cos - sv.z) * (c_cos - sv.z));
        atomicAdd(&part[3], (c_dic - sv.w) * (c_dic - sv.w));
    }
    __syncthreads();
    if (tid < 4) atomicAdd(&out[tid], part[tid]);
}

// ---------------------------------------------------------------------------
extern "C" void kernel_launch(void* const* d_in, const int* in_sizes, int n_in,
                              void* d_out, int out_size, void* d_ws, size_t ws_size,
                              hipStream_t stream) {
    const int*   S    = (const int*)  d_in[0];
    const int*   Mm   = (const int*)  d_in[1];
    const int*   inst = (const int*)  d_in[2];
    const float* sims = (const float*)d_in[3];
    const float* Xc   = (const float*)d_in[4];
    const float* Xr   = (const float*)d_in[5];
    const float* Ac   = (const float*)d_in[6];
    const float* Ar   = (const float*)d_in[7];
    float*       out  = (float*)      d_out;

    // workspace layout (floats): att_c | att_r | emb_c | emb_r | bv  (~17.6 MB)
    const size_t ATT_PAD = 100352;            // NUM_ITEM padded to 256 mult.
    float* ws    = (float*)d_ws;
    float* att_c = ws;
    float* att_r = att_c + ATT_PAD;
    float* emb_c = att_r + ATT_PAD;
    float* emb_r = emb_c + (size_t)NUM_SETS * DIM;
    float* bvp   = emb_r + (size_t)NUM_SETS * DIM;

    hipMemsetAsync(out, 0, (size_t)out_size * sizeof(float), stream);

    // 100000 rows / 16 = 6250 waves; 8 waves per 256-thread block -> 782 blocks
    const int gemv_blocks = (NUM_ITEM / 16 + 7) / 8;
    attn_matvec_wmma<<<gemv_blocks, 256, 0, stream>>>(Xc, Ac, att_c, NUM_ITEM);
    attn_matvec_wmma<<<gemv_blocks, 256, 0, stream>>>(Xr, Ar, att_r, NUM_ITEM);

    set_embed_kernel<<<NUM_SETS, 256, 0, stream>>>(S, Mm, Xc, Xr, att_c, att_r,
                                                   emb_c, emb_r, bvp);

    batch_loss_kernel<<<BATCH / 8, 256, 0, stream>>>(inst, sims, emb_c, emb_r,
                                                     bvp, out);
}